// LNon_72808285601882
// MI455X (gfx1250) — compile-verified
//
#include <hip/hip_runtime.h>
#include <stdint.h>

#define GROUPS 8
#define POINTS 5
#define Bn 64
#define Cn 64
#define HWn 12544              // 112*112
#define CHWn (Cn*HWn)          // 802816
#define CPG (Cn/GROUPS)        // 8
#define NPG (Bn*CPG*HWn)       // 6422528 elements per group
#define TILE_S 256
#define TILES_PER_ROW (HWn/TILE_S)   // 49
#define ROW_PAD 264            // 256 + 8 dwords; 264 % 64 == 8 -> conflict-free banks
#define TAB_STRIDE 24

typedef unsigned int u32x4 __attribute__((ext_vector_type(4)));
typedef int          i32x8 __attribute__((ext_vector_type(8)));
typedef int          i32x4 __attribute__((ext_vector_type(4)));

struct Ws {
  unsigned minE[GROUPS];
  unsigned maxE[GROUPS];
  unsigned counts[GROUPS * POINTS];   // 40
  unsigned pad_[8];
  float    tabs[GROUPS * TAB_STRIDE]; // per group: grid0,step,inv,pad, accum[5], slope[4], ftheta[5], fvelo[5], pad
};

__device__ __forceinline__ unsigned encf(float f) {
  unsigned u = __float_as_uint(f);
  return (u & 0x80000000u) ? ~u : (u | 0x80000000u);  // order-preserving map
}
__device__ __forceinline__ float decf(unsigned u) {
  u = (u & 0x80000000u) ? (u & 0x7FFFFFFFu) : ~u;
  return __uint_as_float(u);
}

__global__ void k_init(Ws* ws) {
  int t = threadIdx.x;
  if (t < GROUPS) { ws->minE[t] = 0xFFFFFFFFu; ws->maxE[t] = 0u; }
  if (t < GROUPS * POINTS) ws->counts[t] = 0u;
}

// One block per (b,c): group = c % 8 is block-uniform; fully coalesced float4 streaming.
__global__ void k_minmax(const float* __restrict__ x, const float* __restrict__ ctp, Ws* ws) {
  __shared__ float smin[256], smax[256];
  const float ct = ctp[0];
  const int bid = blockIdx.x;        // b*64 + c
  const int g = bid & 7;             // c % 8
  const int tid = threadIdx.x;
  const float4* p = (const float4*)(x + (size_t)bid * HWn);
  float lmin = 3.402823466e38f, lmax = -3.402823466e38f;
  for (int i = tid; i < HWn / 4; i += 256) {
    float4 v = p[i];
    float a = v.x * ct, b = v.y * ct, c = v.z * ct, d = v.w * ct;
    lmin = fminf(lmin, fminf(fminf(a, b), fminf(c, d)));
    lmax = fmaxf(lmax, fmaxf(fmaxf(a, b), fmaxf(c, d)));
  }
  smin[tid] = lmin; smax[tid] = lmax;
  __syncthreads();
  for (int s = 128; s > 0; s >>= 1) {
    if (tid < s) {
      smin[tid] = fminf(smin[tid], smin[tid + s]);
      smax[tid] = fmaxf(smax[tid], smax[tid + s]);
    }
    __syncthreads();
  }
  if (tid == 0) {
    atomicMin(&ws->minE[g], encf(smin[0]));
    atomicMax(&ws->maxE[g], encf(smax[0]));
  }
}

__global__ void k_hist(const float* __restrict__ x, const float* __restrict__ ctp, Ws* ws) {
  __shared__ unsigned h[POINTS];
  const float ct = ctp[0];
  const int bid = blockIdx.x, g = bid & 7, tid = threadIdx.x;
  if (tid < POINTS) h[tid] = 0u;
  __syncthreads();
  const float dmin = decf(ws->minE[g]);
  const float dmax = decf(ws->maxE[g]);
  const float scale = (float)POINTS / (dmax - dmin);
  const float4* p = (const float4*)(x + (size_t)bid * HWn);
  unsigned c0 = 0, c1 = 0, c2 = 0, c3 = 0, c4 = 0;
  for (int i = tid; i < HWn / 4; i += 256) {
    float4 v = p[i];
    float e[4] = {v.x * ct, v.y * ct, v.z * ct, v.w * ct};
#pragma unroll
    for (int u = 0; u < 4; ++u) {
      int b = (int)floorf((e[u] - dmin) * scale);
      b = min(max(b, 0), POINTS - 1);
      c0 += (b == 0); c1 += (b == 1); c2 += (b == 2); c3 += (b == 3); c4 += (b == 4);
    }
  }
  if (c0) atomicAdd(&h[0], c0);
  if (c1) atomicAdd(&h[1], c1);
  if (c2) atomicAdd(&h[2], c2);
  if (c3) atomicAdd(&h[3], c3);
  if (c4) atomicAdd(&h[4], c4);
  __syncthreads();
  if (tid < POINTS) atomicAdd(&ws->counts[g * POINTS + tid], h[tid]);
}

// Tiny per-group table build: CDF, slopes, and theta/velo resampled along the CDF.
__global__ void k_tables(const float* __restrict__ params, Ws* ws) {
  int g = threadIdx.x;
  if (g >= GROUPS) return;
  float accum[POINTS];
  const float nInv = 1.0f / (float)NPG;
  const float dmin = decf(ws->minE[g]);
  const float dmax = decf(ws->maxE[g]);
  const float width = dmax - dmin;
  float cum = 0.0f;
  for (int i = 0; i < POINTS; ++i) {
    cum += (float)ws->counts[g * POINTS + i] * nInv;
    accum[i] = cum * (float)(POINTS - 1);
  }
  const float step = width / (float)POINTS;
  const float grid0 = dmin + 0.5f * step;
  float* tb = &ws->tabs[g * TAB_STRIDE];
  tb[0] = grid0; tb[1] = step; tb[2] = 1.0f / step; tb[3] = 0.0f;
  for (int i = 0; i < POINTS; ++i) tb[4 + i] = accum[i];
  for (int i = 0; i < POINTS - 1; ++i) tb[9 + i] = (accum[i + 1] - accum[i]) / step;
  const float* th = params + g * POINTS;                   // params[0][g][:]
  const float* ve = params + GROUPS * POINTS + g * POINTS; // params[1][g][:]
  for (int q = 0; q < POINTS; ++q) {
    float qf = (float)q;
    int i = 0;
    for (int k = 1; k <= POINTS - 2; ++k) if (accum[k] <= qf) i = k;  // searchsorted-right - 1, clipped
    float t = (qf - accum[i]) / (accum[i + 1] - accum[i]);
    tb[13 + q] = th[i] + t * (th[i + 1] - th[i]);
    tb[18 + q] = ve[i] + t * (ve[i + 1] - ve[i]);
  }
}

// TDM-staged transform: one tensor_load_to_lds per block pulls an 8x256 f32 tile
// (row stride HW) into LDS with 8-dword row padding, then threads read transposed
// (conflict-free) and write the permuted output fully coalesced.
__global__ void __launch_bounds__(256) k_transform(
    const float* __restrict__ x, const float* __restrict__ ctp,
    const float* __restrict__ spp, const Ws* __restrict__ ws,
    float* __restrict__ out) {
  __shared__ float tile[GROUPS * ROW_PAD];       // 8 rows, padded stride 264
  __shared__ float tabs[GROUPS * TAB_STRIDE];
  const int tid = threadIdx.x;
  const int bid = blockIdx.x;
  const int b = bid / (CPG * TILES_PER_ROW);
  const int r = bid % (CPG * TILES_PER_ROW);
  const int k = r / TILES_PER_ROW;
  const int t = r % TILES_PER_ROW;
  const int s0 = t * TILE_S;

  if (tid < GROUPS * TAB_STRIDE) tabs[tid] = ws->tabs[tid];

  const float* src = x + (size_t)b * CHWn + (size_t)(k * GROUPS) * HWn + s0;
  if (tid < 32) {  // wave 0 issues the DMA (EXEC ignored by TDM, branch keeps it single-issue)
    unsigned lds_addr = (unsigned)(uintptr_t)(&tile[0]);
    unsigned long long ga = (unsigned long long)(uintptr_t)src;
    u32x4 g0; i32x8 g1; i32x4 gz4; i32x8 gz8;
    // D# group0: count=1, lds_addr, global_addr[56:0], type=2
    g0[0] = 1u;
    g0[1] = lds_addr;
    g0[2] = (unsigned)(ga & 0xFFFFFFFFu);
    g0[3] = (unsigned)((ga >> 32) & 0x01FFFFFFu) | (2u << 30);
    // D# group1: data_size=4B(2), pad_enable=1, pad_interval=7(256dw), pad_amount=7(8dw)
    g1[0] = (int)((2u << 16) | (1u << 20) | (7u << 22) | (7u << 25));
    g1[1] = (int)((unsigned)TILE_S << 16);          // tensor_dim0[15:0] in [31:16]
    g1[2] = (int)((unsigned)GROUPS << 16);          // tensor_dim0 hi=0, tensor_dim1[15:0]=8
    g1[3] = (int)((unsigned)TILE_S << 16);          // tensor_dim1 hi=0, tile_dim0=256
    g1[4] = GROUPS;                                 // tile_dim1=8, tile_dim2=0
    g1[5] = HWn;                                    // tensor_dim0_stride[31:0]
    g1[6] = 0;                                      // stride0 hi, stride1 lo
    g1[7] = 0;
    gz4[0] = 0; gz4[1] = 0; gz4[2] = 0; gz4[3] = 0; // groups 2/3 unused (2D tile)
    gz8[0] = 0; gz8[1] = 0; gz8[2] = 0; gz8[3] = 0;
    gz8[4] = 0; gz8[5] = 0; gz8[6] = 0; gz8[7] = 0;
    __builtin_amdgcn_tensor_load_to_lds(g0, g1, gz4, gz4, gz8, 0);
    __builtin_amdgcn_s_wait_tensorcnt(0);
  }
  __syncthreads();

  const float ct = ctp[0];
  const float sp = spp[0];
  float* dst = out + (size_t)b * CHWn + (size_t)(k * HWn + s0) * GROUPS;
#pragma unroll
  for (int it = 0; it < (GROUPS * TILE_S) / 256; ++it) {
    int j = tid + it * 256;          // output-contiguous index: j = s*8 + ix
    int ix = j & 7;
    int s = j >> 3;
    float xv = tile[ix * ROW_PAD + s] * ct;
    const float* tb = &tabs[ix * TAB_STRIDE];
    float grid0 = tb[0], step = tb[1], inv = tb[2];
    int i = (int)floorf((xv - grid0) * inv);
    i = min(max(i, 0), POINTS - 2);
    float index = tb[4 + i] + (xv - (grid0 + (float)i * step)) * tb[9 + i];
    int bg = (int)floorf(index);
    bg = min(max(bg, 0), POINTS - 1);
    float pos = index - (float)bg;
    int en = min(bg + 1, POINTS - 1);
    float theta = tb[13 + bg] + pos * (tb[13 + en] - tb[13 + bg]);
    float velo  = tb[18 + bg] + pos * (tb[18 + en] - tb[18 + bg]);
    float ds = velo * 0.01f;
    float res = xv * (1.0f + ds * __sinf(theta)) + ds * __cosf(theta);
    __builtin_nontemporal_store(res * sp, &dst[j]);
  }
}

extern "C" void kernel_launch(void* const* d_in, const int* in_sizes, int n_in,
                              void* d_out, int out_size, void* d_ws, size_t ws_size,
                              hipStream_t stream) {
  const float* data   = (const float*)d_in[0];
  const float* params = (const float*)d_in[1];
  const float* ct     = (const float*)d_in[2];
  const float* sp     = (const float*)d_in[3];
  float* out = (float*)d_out;
  Ws* ws = (Ws*)d_ws;

  k_init<<<1, 64, 0, stream>>>(ws);
  k_minmax<<<Bn * Cn, 256, 0, stream>>>(data, ct, ws);
  k_hist<<<Bn * Cn, 256, 0, stream>>>(data, ct, ws);
  k_tables<<<1, 32, 0, stream>>>(params, ws);
  k_transform<<<Bn * CPG * TILES_PER_ROW, 256, 0, stream>>>(data, ct, sp, ws, out);
}